// GraphSageLayer_47201690583087
// MI455X (gfx1250) — compile-verified
//
#include <hip/hip_runtime.h>
#include <hip/hip_bf16.h>

// ---------- CDNA5 WMMA types ----------
typedef float v2f __attribute__((ext_vector_type(2)));
typedef float v8f __attribute__((ext_vector_type(8)));

#define D_FEAT 128

// ===================================================================
// Kernel 1: zero workspace (agg, cnt, bn accumulators)
// ===================================================================
__global__ void zero_ws_kernel(float4* __restrict__ p, int n4) {
    int i = blockIdx.x * blockDim.x + threadIdx.x;
    if (i < n4) p[i] = make_float4(0.f, 0.f, 0.f, 0.f);
}

// ===================================================================
// Kernel 2: edge aggregation.
//  - Each wave32 grabs 32 edges with ONE coalesced index load per lane
//    (src and dst), then iterates the 32 edges, broadcasting indices
//    with __shfl.
//  - Per edge: 512B coalesced float4 gather of x[src] (L2-resident),
//    128 global_atomic_add_f32 into agg[dst] (L2-resident), 1 count
//    atomic. Prefetch next source row to hide gather latency behind
//    the atomic stream.
// ===================================================================
__global__ __launch_bounds__(256) void edge_agg_kernel(
        const float* __restrict__ x,
        const int* __restrict__ ei,   // [2,E] flat
        float* __restrict__ agg,
        float* __restrict__ cnt,
        int E) {
    const int lane  = threadIdx.x & 31;
    const int wave  = (blockIdx.x * blockDim.x + threadIdx.x) >> 5;
    const int nwave = (gridDim.x * blockDim.x) >> 5;

    for (int base = wave * 32; base < E; base += nwave * 32) {
        const int myE = base + lane;
        int sL = 0, dL = -1;
        if (myE < E) {
            sL = ei[myE];        // coalesced: 32 consecutive src indices
            dL = ei[E + myE];    // coalesced: 32 consecutive dst indices
        }
        const int m = min(32, E - base);
        for (int j = 0; j < m; ++j) {
            const int s = __shfl(sL, j, 32);
            const int d = __shfl(dL, j, 32);
            if (j + 1 < m) {   // prefetch next source row into near cache
                const int sn = __shfl(sL, j + 1, 32);
                __builtin_prefetch(x + (size_t)sn * D_FEAT + lane * 4, 0, 3);
            }
            const float4 v = *(const float4*)(x + (size_t)s * D_FEAT + lane * 4);
            float* dp = agg + (size_t)d * D_FEAT + lane * 4;
            atomicAdd(dp + 0, v.x);
            atomicAdd(dp + 1, v.y);
            atomicAdd(dp + 2, v.z);
            atomicAdd(dp + 3, v.w);
            if (lane == 0) atomicAdd(cnt + d, 1.0f);
        }
    }
}

// ===================================================================
// Kernel 3: fused SAGE GEMM + bias + ReLU + BN-stat accumulation.
// Block = 256 threads (8 waves) owns one 16-row tile of the output.
// Wave w owns output columns [16w, 16w+16).
// h = relu(mean @ Wl^T + x @ Wr^T + b), written into d_out (scratch).
// TWO independent WMMA accumulator chains (mean-chain and x-chain) so
// consecutive v_wmma_f32_16x16x4_f32 ops have no RAW dependency.
// ===================================================================
__global__ __launch_bounds__(256) void sage_gemm_kernel(
        const float* __restrict__ x,
        const float* __restrict__ agg,
        const float* __restrict__ cnt,
        const float* __restrict__ Wl,
        const float* __restrict__ Wr,
        const float* __restrict__ bias,
        float* __restrict__ h,          // = d_out used as scratch
        float* __restrict__ bnSum,
        float* __restrict__ bnSq,
        int N) {
    __shared__ float sMean[16 * D_FEAT];
    __shared__ float sX[16 * D_FEAT];
    __shared__ float sSum[D_FEAT];
    __shared__ float sSq[D_FEAT];

    const int tid     = threadIdx.x;
    const int rowBase = blockIdx.x * 16;

    // Stage A tiles into LDS: mean = agg / max(cnt,1), plus raw x.
    for (int i = tid; i < 16 * D_FEAT; i += 256) {
        int r  = i >> 7;
        int c  = i & (D_FEAT - 1);
        int gr = rowBase + r;
        float xv = 0.f, mv = 0.f;
        if (gr < N) {
            xv = x[(size_t)gr * D_FEAT + c];
            mv = agg[(size_t)gr * D_FEAT + c] / fmaxf(cnt[gr], 1.0f);
        }
        sX[i]    = xv;
        sMean[i] = mv;
    }
    if (tid < D_FEAT) { sSum[tid] = 0.f; sSq[tid] = 0.f; }
    __syncthreads();

    const int wave    = tid >> 5;          // 0..7 -> column tile
    const int lane    = tid & 31;
    const int colBase = wave * 16;
    const int lo      = lane & 15;
    const int hi      = lane >> 4;         // 0 or 1 -> K offset +2

    const float* WlRow = Wl + (size_t)(colBase + lo) * D_FEAT; // B[k][n]=Wl[n][k]
    const float* WrRow = Wr + (size_t)(colBase + lo) * D_FEAT;

    v8f accM = {};   // mean @ Wl^T chain
    v8f accX = {};   // x    @ Wr^T chain  (independent of accM)
    #pragma unroll 4
    for (int k0 = 0; k0 < D_FEAT; k0 += 4) {
        const int kk = k0 + 2 * hi;
        v2f aM, aX, bL, bR;
        aM[0] = sMean[lo * D_FEAT + kk];
        aM[1] = sMean[lo * D_FEAT + kk + 1];
        aX[0] = sX[lo * D_FEAT + kk];
        aX[1] = sX[lo * D_FEAT + kk + 1];
        bL[0] = WlRow[kk];
        bL[1] = WlRow[kk + 1];
        bR[0] = WrRow[kk];
        bR[1] = WrRow[kk + 1];
        accM = __builtin_amdgcn_wmma_f32_16x16x4_f32(
                false, aM, false, bL, (short)0, accM, false, false);
        accX = __builtin_amdgcn_wmma_f32_16x16x4_f32(
                false, aX, false, bR, (short)0, accX, false, false);
    }

    const float bj = bias[colBase + lo];
    float ps = 0.f, psq = 0.f;

    // N is a multiple of 16 in this workload -> all tiles full; keep a
    // single uniform branch for generality instead of per-element guards.
    float* hRow = h + (size_t)(rowBase + 8 * hi) * D_FEAT + colBase + lo;
    if (rowBase + 16 <= N) {
        #pragma unroll
        for (int v = 0; v < 8; ++v) {
            float hv = fmaxf(accM[v] + accX[v] + bj, 0.f);
            hRow[(size_t)v * D_FEAT] = hv;
            ps  += hv;
            psq += hv * hv;
        }
    } else {
        #pragma unroll
        for (int v = 0; v < 8; ++v) {
            float hv = fmaxf(accM[v] + accX[v] + bj, 0.f);
            int gr = rowBase + v + 8 * hi;
            if (gr < N) hRow[(size_t)v * D_FEAT] = hv; else hv = 0.f;
            ps  += hv;
            psq += hv * hv;
        }
    }

    // per-block column reduction in LDS (ds_add_f32), then 1 global atomic/col
    atomicAdd(&sSum[colBase + lo], ps);
    atomicAdd(&sSq[colBase + lo], psq);
    __syncthreads();
    if (tid < D_FEAT) {
        atomicAdd(&bnSum[tid], sSum[tid]);
        atomicAdd(&bnSq[tid],  sSq[tid]);
    }
}

// ===================================================================
// Kernel 4: fold BN stats into per-column affine scale/shift.
// ===================================================================
__global__ void bn_finalize_kernel(const float* __restrict__ bnSum,
                                   const float* __restrict__ bnSq,
                                   const float* __restrict__ gamma,
                                   const float* __restrict__ beta,
                                   float* __restrict__ scale,
                                   float* __restrict__ shift,
                                   float invN) {
    int c = threadIdx.x;
    float mu  = bnSum[c] * invN;
    float var = bnSq[c] * invN - mu * mu;
    float s   = gamma[c] * rsqrtf(var + 1e-5f);
    scale[c] = s;
    shift[c] = beta[c] - mu * s;
}

// ===================================================================
// Kernel 5: out = x + h*scale + shift  (h lives in d_out, in-place)
// ===================================================================
__global__ void residual_kernel(const float* __restrict__ x,
                                const float* __restrict__ scale,
                                const float* __restrict__ shift,
                                float* __restrict__ out,
                                int n4) {
    int i = blockIdx.x * blockDim.x + threadIdx.x;
    if (i >= n4) return;
    int c4 = i & ((D_FEAT / 4) - 1);           // which float4 of the row
    float4 h4 = ((const float4*)out)[i];
    float4 x4 = ((const float4*)x)[i];
    float4 sc = ((const float4*)scale)[c4];
    float4 sh = ((const float4*)shift)[c4];
    float4 r;
    r.x = x4.x + h4.x * sc.x + sh.x;
    r.y = x4.y + h4.y * sc.y + sh.y;
    r.z = x4.z + h4.z * sc.z + sh.z;
    r.w = x4.w + h4.w * sc.w + sh.w;
    ((float4*)out)[i] = r;
}

// ===================================================================
extern "C" void kernel_launch(void* const* d_in, const int* in_sizes, int n_in,
                              void* d_out, int out_size, void* d_ws, size_t ws_size,
                              hipStream_t stream) {
    const float* x     = (const float*)d_in[0];
    const int*   ei    = (const int*)  d_in[1];   // [2,E] flat
    const float* Wl    = (const float*)d_in[2];
    const float* Wr    = (const float*)d_in[3];
    const float* bias  = (const float*)d_in[4];
    const float* gamma = (const float*)d_in[5];
    const float* beta  = (const float*)d_in[6];
    float* out = (float*)d_out;

    const int N = in_sizes[0] / D_FEAT;           // 50000
    const int E = in_sizes[1] / 2;                // 800000

    // ---- workspace layout (floats), contiguous ----
    float* wsF   = (float*)d_ws;
    float* agg   = wsF;                            // N*128
    float* cnt   = agg + (size_t)N * D_FEAT;       // N
    float* bnSum = cnt + N;                        // 128
    float* bnSq  = bnSum + D_FEAT;                 // 128
    float* scale = bnSq + D_FEAT;                  // 128
    float* shift = scale + D_FEAT;                 // 128

    // 1) zero agg + cnt + bnSum + bnSq
    {
        size_t zf = (size_t)N * D_FEAT + (size_t)N + 2 * D_FEAT;
        int n4 = (int)((zf + 3) / 4);
        int blocks = (n4 + 255) / 256;
        zero_ws_kernel<<<blocks, 256, 0, stream>>>((float4*)d_ws, n4);
    }
    // 2) edge scatter-aggregate: one sweep = 3125 blocks * 8 waves * 32 edges
    {
        int waves  = (E + 31) / 32;               // 25000
        int blocks = (waves + 7) / 8;             // 3125
        edge_agg_kernel<<<blocks, 256, 0, stream>>>(x, ei, agg, cnt, E);
    }
    // 3) fused GEMM (WMMA f32) + bias + relu + BN partials; h -> d_out
    {
        int blocks = (N + 15) / 16;
        sage_gemm_kernel<<<blocks, 256, 0, stream>>>(
            x, agg, cnt, Wl, Wr, bias, out, bnSum, bnSq, N);
    }
    // 4) BN scale/shift
    bn_finalize_kernel<<<1, D_FEAT, 0, stream>>>(
        bnSum, bnSq, gamma, beta, scale, shift, 1.0f / (float)N);
    // 5) residual + affine, in-place on d_out
    {
        int n4 = (N * D_FEAT) / 4;
        int blocks = (n4 + 255) / 256;
        residual_kernel<<<blocks, 256, 0, stream>>>(x, scale, shift, out, n4);
    }
}